// MS_GWCN_77369540870373
// MI455X (gfx1250) — compile-verified
//
#include <hip/hip_runtime.h>

typedef float v2f __attribute__((ext_vector_type(2)));
typedef float v8f __attribute__((ext_vector_type(8)));

#define N_NODES 25000
#define E2      400000   // directed edges

// ---------------- CSR construction ----------------

__global__ void msgwcn_deg_kernel(const int* __restrict__ row, int* __restrict__ deg, int n) {
    int i = blockIdx.x * blockDim.x + threadIdx.x;
    if (i < n) atomicAdd(&deg[row[i]], 1);
}

__global__ void msgwcn_dinv_kernel(const int* __restrict__ deg, float* __restrict__ dinv, int n) {
    int i = blockIdx.x * blockDim.x + threadIdx.x;
    if (i < n) {
        int d = deg[i];
        dinv[i] = (d > 0) ? rsqrtf((float)d) : 0.0f;
    }
}

// exclusive prefix sum over N entries, single workgroup (1024 threads)
__global__ void msgwcn_scan_kernel(const int* __restrict__ deg, int* __restrict__ rp, int n) {
    __shared__ int s[1024];
    __shared__ int carry;
    int tid = threadIdx.x;
    if (tid == 0) carry = 0;
    __syncthreads();
    for (int base = 0; base < n; base += 1024) {
        int i = base + tid;
        int v = (i < n) ? deg[i] : 0;
        s[tid] = v;
        __syncthreads();
        for (int off = 1; off < 1024; off <<= 1) {
            int t = (tid >= off) ? s[tid - off] : 0;
            __syncthreads();
            s[tid] += t;
            __syncthreads();
        }
        int excl = s[tid] - v;   // exclusive within chunk
        if (i < n) rp[i] = carry + excl;
        __syncthreads();
        if (tid == 1023) carry += s[1023];
        __syncthreads();
    }
    if (tid == 0) rp[n] = carry;
}

__global__ void msgwcn_fill_kernel(const int* __restrict__ row, const int* __restrict__ col,
                                   const float* __restrict__ dinv, const float* __restrict__ lam,
                                   const int* __restrict__ rp, int* __restrict__ cursor,
                                   int* __restrict__ ccol, float* __restrict__ cw, int n) {
    int i = blockIdx.x * blockDim.x + threadIdx.x;
    if (i < n) {
        int r = row[i];
        int c = col[i];
        int pos = rp[r] + atomicAdd(&cursor[r], 1);
        float s = 2.0f / lam[0];
        ccol[pos] = c;
        cw[pos]   = -s * dinv[r] * dinv[c];
    }
}

// ---------------- Chebyshev propagation (CSR gather SpMM) ----------------
// dst[n,:] = alpha * (diag*a[n,:] + sum_e w[e]*a[col[e],:]) - (subFlag ? bprev[n,:] : 0)
__global__ void msgwcn_prop_kernel(float* __restrict__ dst, const float* __restrict__ a,
                                   const float* __restrict__ bprev, const int* __restrict__ rp,
                                   const int* __restrict__ ccol, const float* __restrict__ cw,
                                   const float* __restrict__ lam, int c4count,
                                   float alpha, int subFlag) {
    int idx = blockIdx.x * blockDim.x + threadIdx.x;
    int total = N_NODES * c4count;
    if (idx >= total) return;
    int n  = idx / c4count;
    int c4 = idx - n * c4count;
    int C  = c4count * 4;
    float diag = 2.0f / lam[0] - 1.0f;

    const float4 av = *(const float4*)(a + (size_t)n * C + c4 * 4);
    float ax = diag * av.x, ay = diag * av.y, az = diag * av.z, aw = diag * av.w;

    int beg = rp[n], end = rp[n + 1];
    for (int e = beg; e < end; ++e) {
        int   c = ccol[e];
        float w = cw[e];
        const float4 hv = *(const float4*)(a + (size_t)c * C + c4 * 4);
        ax += w * hv.x; ay += w * hv.y; az += w * hv.z; aw += w * hv.w;
    }
    ax *= alpha; ay *= alpha; az *= alpha; aw *= alpha;
    if (subFlag) {
        const float4 bv = *(const float4*)(bprev + (size_t)n * C + c4 * 4);
        ax -= bv.x; ay -= bv.y; az -= bv.z; aw -= bv.w;
    }
    *(float4*)(dst + (size_t)n * C + c4 * 4) = make_float4(ax, ay, az, aw);
}

// ---------------- fp32 WMMA GEMM: C[n, col] (+)= A[n, k] @ B[k, col] ----------------
// blockDim = 128 (4 waves); each wave owns a 16 x (16*NT) output tile.
// A fragment is shared across the NT column tiles (halves A traffic).
// Inner loop carries incremented pointers (no per-iteration 64-bit address math)
// and uses clamped-row loads + select instead of exec-divergent guarded loads.
// initBias: acc = bias[col] (first Chebyshev term) else acc = C (accumulate).
// relu: applied on final store (last Chebyshev term of this scale).
template <int NT>
__global__ void msgwcn_gemm_wmma_f32(const float* __restrict__ A, int lda,
                                     const float* __restrict__ B, int ldb,
                                     const float* __restrict__ bias,
                                     float* __restrict__ C, int ldc,
                                     int nrows, int kdim, int initBias, int relu) {
    int wave = threadIdx.x >> 5;
    int lane = threadIdx.x & 31;
    int half = lane >> 4;     // 0: K+0/K+1, rows M+0..7 ; 1: K+2/K+3, rows M+8..15
    int l16  = lane & 15;
    int tileM = (blockIdx.x * 4 + wave) * 16;
    int col0  = blockIdx.y * (16 * NT) + l16;

    v8f acc[NT];
#pragma unroll
    for (int t = 0; t < NT; ++t) {
        if (initBias) {
            float bv = bias[col0 + t * 16];
#pragma unroll
            for (int j = 0; j < 8; ++j) acc[t][j] = bv;
        } else {
#pragma unroll
            for (int j = 0; j < 8; ++j) {
                int r = tileM + j + half * 8;
                acc[t][j] = (r < nrows) ? C[(size_t)r * ldc + col0 + t * 16] : 0.0f;
            }
        }
    }

    int  arow = tileM + l16;
    bool aok  = arow < nrows;
    int  crow = aok ? arow : 0;                       // clamped: always a valid address
    const float* Ap  = A + (size_t)crow * lda + half * 2;
    const float* Bp0 = B + (size_t)(half * 2) * ldb + col0;
    const float* Bp1 = Bp0 + ldb;
    const int bstep  = 4 * ldb;

#pragma unroll 4
    for (int kk = 0; kk < kdim; kk += 4) {
        const float2 t2 = *(const float2*)Ap;         // unconditional load
        v2f a;
        a.x = aok ? t2.x : 0.0f;                      // v_cndmask, no exec games
        a.y = aok ? t2.y : 0.0f;
        v2f b0;
        b0.x = Bp0[0];
        b0.y = Bp1[0];
        acc[0] = __builtin_amdgcn_wmma_f32_16x16x4_f32(false, a, false, b0, (short)0,
                                                       acc[0], false, false);
        if (NT == 2) {
            v2f b1;
            b1.x = Bp0[16];                           // immediate-offset loads
            b1.y = Bp1[16];
            acc[1] = __builtin_amdgcn_wmma_f32_16x16x4_f32(false, a, false, b1, (short)0,
                                                           acc[1], false, false);
        }
        Ap  += 4;
        Bp0 += bstep;
        Bp1 += bstep;
    }

#pragma unroll
    for (int t = 0; t < NT; ++t) {
#pragma unroll
        for (int j = 0; j < 8; ++j) {
            int r = tileM + j + half * 8;
            if (r < nrows) {
                float v = acc[t][j];
                if (relu) v = fmaxf(v, 0.0f);
                C[(size_t)r * ldc + col0 + t * 16] = v;
            }
        }
    }
}

// ---------------- driver ----------------

extern "C" void kernel_launch(void* const* d_in, const int* in_sizes, int n_in,
                              void* d_out, int out_size, void* d_ws, size_t ws_size,
                              hipStream_t stream) {
    (void)in_sizes; (void)n_in; (void)out_size; (void)ws_size;

    const float* x    = (const float*)d_in[0];
    const int*   edge = (const int*)d_in[1];     // [2, 2E] int32
    const float* lam  = (const float*)d_in[2];
    const int*   erow = edge;
    const int*   ecol = edge + E2;

    char*  ws  = (char*)d_ws;
    size_t off = 0;
    auto carve = [&](size_t bytes) -> void* {
        void* p = ws + off;
        off += (bytes + 255) & ~(size_t)255;
        return p;
    };
    int*   deg    = (int*)carve((size_t)N_NODES * 4);
    int*   cursor = (int*)carve((size_t)N_NODES * 4);
    float* dinv   = (float*)carve((size_t)N_NODES * 4);
    int*   rp     = (int*)carve((size_t)(N_NODES + 1) * 4);
    int*   ccol   = (int*)carve((size_t)E2 * 4);
    float* cw     = (float*)carve((size_t)E2 * 4);
    const size_t bufBytes = (size_t)N_NODES * 768 * 4;
    float* bufA = (float*)carve(bufBytes);
    float* bufB = (float*)carve(bufBytes);
    float* bufC = (float*)carve(bufBytes);
    float* bufD = (float*)carve(bufBytes);

    // graph preprocessing: degree -> dinv -> row_ptr -> CSR(col, w)
    hipMemsetAsync(deg,    0, (size_t)N_NODES * 4, stream);
    hipMemsetAsync(cursor, 0, (size_t)N_NODES * 4, stream);
    msgwcn_deg_kernel <<<(E2 + 255) / 256, 256, 0, stream>>>(erow, deg, E2);
    msgwcn_dinv_kernel<<<(N_NODES + 255) / 256, 256, 0, stream>>>(deg, dinv, N_NODES);
    msgwcn_scan_kernel<<<1, 1024, 0, stream>>>(deg, rp, N_NODES);
    msgwcn_fill_kernel<<<(E2 + 255) / 256, 256, 0, stream>>>(erow, ecol, dinv, lam, rp,
                                                             cursor, ccol, cw, E2);

    // layer 0 input: copy x into activation buffer (T0 storage gets recycled at k=3)
    hipMemcpyAsync(bufA, x, (size_t)N_NODES * 128 * 4, hipMemcpyDeviceToDevice, stream);

    const int ins[4] = {128, 192, 384, 768};
    const int hid[4] = {64, 128, 256, 256};
    const int KL[3]  = {2, 4, 6};

    float* H = bufA;   // layer input  (also serves as T0 / T3 storage)
    float* O = bufD;   // layer output (concat of scales)
    for (int li = 0; li < 4; ++li) {
        int Cin = ins[li], Cout = hid[li], C4 = Cin / 4;
        float* Tp[6] = {H, bufB, bufC, H, bufB, bufC};
        dim3 pg((N_NODES * C4 + 255) / 256);
        for (int k = 0; k < 6; ++k) {
            if (k == 1) {
                // T1 = L_hat @ T0
                msgwcn_prop_kernel<<<pg, 256, 0, stream>>>(Tp[1], Tp[0], nullptr, rp, ccol,
                                                           cw, lam, C4, 1.0f, 0);
            } else if (k >= 2) {
                // Tk = 2 * L_hat @ T(k-1) - T(k-2)
                msgwcn_prop_kernel<<<pg, 256, 0, stream>>>(Tp[k], Tp[k - 1], Tp[k - 2], rp,
                                                           ccol, cw, lam, C4, 2.0f, 1);
            }
            for (int si = 0; si < 3; ++si) {
                if (k >= KL[si]) continue;
                const float* W  = (const float*)d_in[3 + 2 * (li * 3 + si)]
                                  + (size_t)k * Cin * Cout;
                const float* bb = (const float*)d_in[4 + 2 * (li * 3 + si)];
                dim3 gg((N_NODES + 63) / 64, Cout / 32);
                msgwcn_gemm_wmma_f32<2><<<gg, 128, 0, stream>>>(
                    Tp[k], Cin, W, Cout, bb, O + si * Cout, 3 * Cout, N_NODES, Cin,
                    (k == 0) ? 1 : 0, (k == KL[si] - 1) ? 1 : 0);
            }
        }
        float* t = H; H = O; O = t;   // next layer input = this layer's (ReLU'd) output
    }

    // final FC: [N,768] @ [768,16] + b  (single 16-wide column tile)
    const float* fcw = (const float*)d_in[27];
    const float* fcb = (const float*)d_in[28];
    dim3 gf((N_NODES + 63) / 64, 1);
    msgwcn_gemm_wmma_f32<1><<<gf, 128, 0, stream>>>(H, 768, fcw, 16, fcb, (float*)d_out, 16,
                                                    N_NODES, 768, 1, 0);
}